// EquiformerV2Layer_36807869727019
// MI455X (gfx1250) — compile-verified
//
#include <hip/hip_runtime.h>
#include <math.h>

#define LTOT 25
#define CCH 64
#define NHEADS 8
#define EPSV 1e-6f

typedef float v2f __attribute__((ext_vector_type(2)));
typedef float v8f __attribute__((ext_vector_type(8)));

__device__ __forceinline__ int deg_of(int l) {
  return (l >= 16) ? 4 : (l >= 9) ? 3 : (l >= 4) ? 2 : (l >= 1) ? 1 : 0;
}

__device__ __forceinline__ float siluf(float x) { return x / (1.f + __expf(-x)); }
__device__ __forceinline__ float sigf(float x)  { return 1.f / (1.f + __expf(-x)); }

__device__ __forceinline__ v8f wmma4(v2f a, v2f b, v8f c) {
  return __builtin_amdgcn_wmma_f32_16x16x4_f32(false, a, false, b, (short)0, c, false, false);
}

// Order-preserving float<->uint for atomicMax-based segment max
__device__ __forceinline__ unsigned f2key(float f) {
  unsigned b = __float_as_uint(f);
  return (b & 0x80000000u) ? ~b : (b | 0x80000000u);
}
__device__ __forceinline__ float key2f(unsigned k) {
  return (k & 0x80000000u) ? __uint_as_float(k & 0x7FFFFFFFu) : __uint_as_float(~k);
}

// ---------------------------------------------------------------------------
// Generic WMMA f32 16x16x4 tile engine.
//   aload(row,k)->float  (row = mt*16 + r, zero-padded by caller)
//   bload(mt,k,col)->float
//   cstore(mt,row,col,acc)
// f32 frag layout (ISA 7.12.2): A 16x4: lanes0-15 M, v0=K0 v1=K1; lanes16-31
// M, v0=K2 v1=K3.  B 4x16 mirrored.  C/D: vgpr r -> M=r+8*(lane>=16), N=lane&15.
// ---------------------------------------------------------------------------
template <class AF, class BF, class CF>
__device__ __forceinline__ void gemm_tiles(int mtiles, int ntiles, int ksteps,
                                           int wid, int nwaves,
                                           AF aload, BF bload, CF cstore) {
  const int lane = threadIdx.x & 31;
  const int half = lane >> 4;
  const int r16  = lane & 15;
  for (int t = wid; t < mtiles * ntiles; t += nwaves) {
    const int mt = t / ntiles;
    const int nt = t % ntiles;
    v8f acc = {0.f, 0.f, 0.f, 0.f, 0.f, 0.f, 0.f, 0.f};
    const int arow = mt * 16 + r16;
    const int ncol = nt * 16 + r16;
    for (int ks = 0; ks < ksteps; ++ks) {
      const int k = ks * 4 + half * 2;
      v2f a, b;
      a.x = aload(arow, k);
      a.y = aload(arow, k + 1);
      b.x = bload(mt, k, ncol);
      b.y = bload(mt, k + 1, ncol);
      acc = wmma4(a, b, acc);
    }
    const int mbase = mt * 16 + half * 8;
#pragma unroll
    for (int vr = 0; vr < 8; ++vr) cstore(mt, mbase + vr, ncol, acc[vr]);
  }
}

// ---------------------------------------------------------------------------
// K0: per-(node,degree) RMS norm (norm1)
// ---------------------------------------------------------------------------
__global__ void k_norm(const float* __restrict__ x_in, const float* __restrict__ w,
                       float* __restrict__ x_out) {
  const int n = blockIdx.x;
  const float* x = x_in + (size_t)n * (LTOT * CCH);
  float acc[5] = {0.f, 0.f, 0.f, 0.f, 0.f};
  for (int idx = threadIdx.x; idx < LTOT * CCH; idx += blockDim.x) {
    float v = x[idx];
    acc[deg_of(idx >> 6)] += v * v;
  }
  __shared__ float part[5][8];
  __shared__ float s_inv[5];
  const int wid = threadIdx.x >> 5, lane = threadIdx.x & 31;
#pragma unroll
  for (int d = 0; d < 5; ++d) {
    float v = acc[d];
    for (int off = 16; off > 0; off >>= 1) v += __shfl_down(v, off, 32);
    if (lane == 0) part[d][wid] = v;
  }
  __syncthreads();
  if (threadIdx.x < 5) {
    float s = 0.f;
    for (int wv = 0; wv < (int)(blockDim.x >> 5); ++wv) s += part[threadIdx.x][wv];
    float cnt = (float)(2 * threadIdx.x + 1) * (float)CCH;
    s_inv[threadIdx.x] = rsqrtf(s / cnt + EPSV);
  }
  __syncthreads();
  float* xo = x_out + (size_t)n * (LTOT * CCH);
  for (int idx = threadIdx.x; idx < LTOT * CCH; idx += blockDim.x) {
    int l = idx >> 6, c = idx & 63;
    int d = deg_of(l);
    xo[idx] = x[idx] * s_inv[d] * w[d * CCH + c];
  }
}

// ---------------------------------------------------------------------------
// K1: edge radial MLP: rad = silu(e@W_r1+b1)@W_r2+b2, e = dist@W_edge + embs
// 64 edges / block, 4 waves, each wave owns one 16-edge M tile.
// ---------------------------------------------------------------------------
__global__ void k_radial(const float* __restrict__ dist, const int* __restrict__ spec,
                         const int* __restrict__ snd, const int* __restrict__ rcv,
                         const float* __restrict__ W_edge, const float* __restrict__ emb_s,
                         const float* __restrict__ emb_d, const float* __restrict__ W_r1,
                         const float* __restrict__ b_r1, const float* __restrict__ W_r2,
                         const float* __restrict__ b_r2, float* __restrict__ rad) {
  __shared__ float sA[64][68];
  __shared__ float sE[64][68];
  __shared__ float sT[64][68];
  const int e0 = blockIdx.x * 64;
  for (int idx = threadIdx.x; idx < 64 * 64; idx += blockDim.x) {
    int e = idx >> 6, c = idx & 63;
    sA[e][c] = dist[(size_t)(e0 + e) * 64 + c];
  }
  __syncthreads();
  const int wid = threadIdx.x >> 5;
  const int we0 = wid * 16;
  gemm_tiles(1, 4, 16, 0, 1,
    [&](int row, int k) { return sA[we0 + row][k]; },
    [&](int, int k, int col) { return W_edge[k * 64 + col]; },
    [&](int, int row, int col, float acc) {
      int e = e0 + we0 + row;
      sE[we0 + row][col] = acc + emb_s[spec[snd[e]] * 64 + col]
                               + emb_d[spec[rcv[e]] * 64 + col];
    });
  __syncthreads();
  gemm_tiles(1, 4, 16, 0, 1,
    [&](int row, int k) { return sE[we0 + row][k]; },
    [&](int, int k, int col) { return W_r1[k * 64 + col]; },
    [&](int, int row, int col, float acc) { sT[we0 + row][col] = siluf(acc + b_r1[col]); });
  __syncthreads();
  gemm_tiles(1, 4, 16, 0, 1,
    [&](int row, int k) { return sT[we0 + row][k]; },
    [&](int, int k, int col) { return W_r2[k * 64 + col]; },
    [&](int, int row, int col, float acc) {
      rad[(size_t)(e0 + we0 + row) * 64 + col] = acc + b_r2[col];
    });
}

// ---------------------------------------------------------------------------
// K2: zero-init accumulators
// ---------------------------------------------------------------------------
__global__ void k_init(float* __restrict__ agg, float* __restrict__ denom,
                       unsigned* __restrict__ nmax, int nAgg, int nND) {
  int i = blockIdx.x * blockDim.x + threadIdx.x;
  int stride = gridDim.x * blockDim.x;
  for (int t = i; t < nAgg; t += stride) agg[t] = 0.f;
  for (int t = i; t < nND; t += stride) { denom[t] = 0.f; nmax[t] = 0u; }
}

// ---------------------------------------------------------------------------
// K3: per-edge: gather + WMMA Wigner rotate + per-degree WMMA vs W_msg, gate,
// attention logits.  msg tensor never touches HBM.
// ---------------------------------------------------------------------------
__global__ void k_edge_msg(const float* __restrict__ xn, const int* __restrict__ snd,
                           const int* __restrict__ rcv, const float* __restrict__ wig,
                           const float* __restrict__ rad, const float* __restrict__ W_msg,
                           const float* __restrict__ W_alpha,
                           const float* __restrict__ alpha_vec,
                           float* __restrict__ h_out, float* __restrict__ logits) {
  __shared__ float s_cat[LTOT][132];
  __shared__ float s_rot[LTOT][132];
  __shared__ float s_wig[LTOT][28];
  __shared__ float s_h[LTOT][68];
  __shared__ float s_rad[64];
  __shared__ float s_sig[64];
  __shared__ float s_a[128];
  const int e = blockIdx.x;
  const int s = snd[e], rr = rcv[e];
  const int wid = threadIdx.x >> 5, nw = blockDim.x >> 5;
  for (int idx = threadIdx.x; idx < LTOT * 128; idx += blockDim.x) {
    int j = idx >> 7, c = idx & 127;
    s_cat[j][c] = (c < 64) ? xn[(size_t)s * (LTOT * 64) + j * 64 + c]
                           : xn[(size_t)rr * (LTOT * 64) + j * 64 + (c - 64)];
  }
  for (int idx = threadIdx.x; idx < LTOT * LTOT; idx += blockDim.x)
    s_wig[idx / 25][idx % 25] = wig[(size_t)e * 625 + idx];
  if (threadIdx.x < 64) s_rad[threadIdx.x] = rad[(size_t)e * 64 + threadIdx.x];
  __syncthreads();
  // WMMA rotate: rot[i][c] = sum_j wig[i][j] * cat[j][c]  (M=25,K=25 -> N=128)
  gemm_tiles(2, 8, 7, wid, nw,
    [&](int row, int k) { return (row < LTOT && k < LTOT) ? s_wig[row][k] : 0.f; },
    [&](int, int k, int col) { return (k < LTOT) ? s_cat[k][col] : 0.f; },
    [&](int, int row, int col, float acc) { if (row < LTOT) s_rot[row][col] = acc; });
  __syncthreads();
  // per-degree GEMM (degrees stacked as M-tiles): h = (rot @ W_msg[d]) * rad
  gemm_tiles(5, 4, 32, wid, nw,
    [&](int row, int k) {
      int d = row >> 4, lm = row & 15;
      return (lm <= 2 * d) ? s_rot[d * d + lm][k] : 0.f;
    },
    [&](int mt, int k, int col) { return W_msg[(size_t)mt * (128 * 64) + k * 64 + col]; },
    [&](int, int row, int col, float acc) {
      int d = row >> 4, lm = row & 15;
      if (lm <= 2 * d) s_h[d * d + lm][col] = acc * s_rad[col];
    });
  __syncthreads();
  if (threadIdx.x < 64) s_sig[threadIdx.x] = sigf(s_h[0][threadIdx.x]);
  __syncthreads();
  for (int idx = threadIdx.x; idx < LTOT * 64; idx += blockDim.x) {
    int l = idx >> 6, c = idx & 63;
    float v = s_h[l][c] * s_sig[c];
    s_h[l][c] = v;
    h_out[(size_t)e * (LTOT * 64) + idx] = v;
  }
  __syncthreads();
  if (threadIdx.x < 128) {
    float acc = 0.f;
    for (int c = 0; c < 64; ++c) acc += s_h[0][c] * W_alpha[c * 128 + threadIdx.x];
    s_a[threadIdx.x] = (acc > 0.f) ? acc : 0.2f * acc;
  }
  __syncthreads();
  if (threadIdx.x < NHEADS) {
    float acc = 0.f;
    for (int a = 0; a < 16; ++a)
      acc += s_a[threadIdx.x * 16 + a] * alpha_vec[threadIdx.x * 16 + a];
    logits[(size_t)e * NHEADS + threadIdx.x] = acc;
  }
}

// ---------------------------------------------------------------------------
// K4/K5: segment softmax over receivers (atomicMax on monotone keys + sum)
// ---------------------------------------------------------------------------
__global__ void k_attn_max(const float* __restrict__ logits, const int* __restrict__ rcv,
                           unsigned* __restrict__ nmax, int E) {
  int idx = blockIdx.x * blockDim.x + threadIdx.x;
  if (idx >= E * NHEADS) return;
  int e = idx >> 3, hh = idx & 7;
  atomicMax(&nmax[rcv[e] * NHEADS + hh], f2key(logits[idx]));
}

__global__ void k_attn_sum(const float* __restrict__ logits, const int* __restrict__ rcv,
                           const unsigned* __restrict__ nmax, float* __restrict__ ex,
                           float* __restrict__ denom, int E) {
  int idx = blockIdx.x * blockDim.x + threadIdx.x;
  if (idx >= E * NHEADS) return;
  int e = idx >> 3, hh = idx & 7;
  float m = key2f(nmax[rcv[e] * NHEADS + hh]);
  float v = __expf(logits[idx] - m);
  ex[idx] = v;
  atomicAdd(&denom[rcv[e] * NHEADS + hh], v);
}

// ---------------------------------------------------------------------------
// K6: per-edge: v = (h @ W_val[deg]) * attn, WMMA rotate-back (wig^T) with
// direct global_atomic_add_f32 scatter epilogue into agg.
// ---------------------------------------------------------------------------
__global__ void k_edge_val(const float* __restrict__ h_in, const int* __restrict__ rcv,
                           const float* __restrict__ wig, const float* __restrict__ W_val,
                           const float* __restrict__ ex, const float* __restrict__ denom,
                           float* __restrict__ agg) {
  __shared__ float s_h[LTOT][68];
  __shared__ float s_v[LTOT][132];
  __shared__ float s_wig[LTOT][28];
  __shared__ float s_attn[NHEADS];
  const int e = blockIdx.x;
  const int rr = rcv[e];
  const int wid = threadIdx.x >> 5, nw = blockDim.x >> 5;
  for (int idx = threadIdx.x; idx < LTOT * 64; idx += blockDim.x)
    s_h[idx >> 6][idx & 63] = h_in[(size_t)e * (LTOT * 64) + idx];
  for (int idx = threadIdx.x; idx < LTOT * LTOT; idx += blockDim.x)
    s_wig[idx / 25][idx % 25] = wig[(size_t)e * 625 + idx];
  if (threadIdx.x < NHEADS)
    s_attn[threadIdx.x] = ex[(size_t)e * NHEADS + threadIdx.x] /
                          (denom[rr * NHEADS + threadIdx.x] + EPSV);
  __syncthreads();
  gemm_tiles(5, 8, 16, wid, nw,
    [&](int row, int k) {
      int d = row >> 4, lm = row & 15;
      return (lm <= 2 * d) ? s_h[d * d + lm][k] : 0.f;
    },
    [&](int mt, int k, int col) { return W_val[(size_t)mt * (64 * 128) + k * 128 + col]; },
    [&](int, int row, int col, float acc) {
      int d = row >> 4, lm = row & 15;
      if (lm <= 2 * d) s_v[d * d + lm][col] = acc * s_attn[col >> 4];
    });
  __syncthreads();
  // WMMA rotate-back: vr[i][c] = sum_j wig[j][i] * v[j][c], scatter-add to agg
  gemm_tiles(2, 8, 7, wid, nw,
    [&](int row, int k) { return (row < LTOT && k < LTOT) ? s_wig[k][row] : 0.f; },
    [&](int, int k, int col) { return (k < LTOT) ? s_v[k][col] : 0.f; },
    [&](int, int row, int col, float acc) {
      if (row < LTOT)
        atomicAdd(&agg[(size_t)rr * (LTOT * 128) + row * 128 + col], acc);
    });
}

// ---------------------------------------------------------------------------
// K7: x1 = deg_linear(agg, W_out) + node_feats; yn = norm2(x1). 2 nodes/block.
// M tiles: degrees 0..3 -> 1 tile (rows 2..14), degree 4 -> 2 tiles (18 rows)
// ---------------------------------------------------------------------------
__global__ void k_node_out(const float* __restrict__ agg, const float* __restrict__ nf,
                           const float* __restrict__ W_out, const float* __restrict__ n2w,
                           float* __restrict__ x1, float* __restrict__ yn) {
  __shared__ float s_agg[2 * LTOT][132];
  __shared__ float s_x1[2 * LTOT][68];
  __shared__ float s_ss[2][5];
  __shared__ float s_inv[2][5];
  const int n0 = blockIdx.x * 2;
  for (int idx = threadIdx.x; idx < 2 * LTOT * 128; idx += blockDim.x)
    s_agg[idx >> 7][idx & 127] = agg[(size_t)n0 * (LTOT * 128) + idx];
  if (threadIdx.x < 10) s_ss[threadIdx.x / 5][threadIdx.x % 5] = 0.f;
  __syncthreads();
  gemm_tiles(6, 4, 32, threadIdx.x >> 5, blockDim.x >> 5,
    [&](int row, int k) {
      int mt = row >> 4;
      int d = (mt < 5) ? mt : 4;
      int rl = ((mt == 5) ? 16 : 0) + (row & 15);
      int span = 2 * d + 1;
      if (rl >= 2 * span) return 0.f;
      int item = (rl >= span) ? 1 : 0;
      int l = d * d + rl - item * span;
      return s_agg[item * LTOT + l][k];
    },
    [&](int mt, int k, int col) {
      int d = (mt < 5) ? mt : 4;
      return W_out[(size_t)d * (128 * 64) + k * 64 + col];
    },
    [&](int mt, int row, int col, float acc) {
      int d = (mt < 5) ? mt : 4;
      int rl = ((mt == 5) ? 16 : 0) + (row & 15);
      int span = 2 * d + 1;
      if (rl >= 2 * span) return;
      int item = (rl >= span) ? 1 : 0;
      int l = d * d + rl - item * span;
      float v = acc + nf[(size_t)(n0 + item) * (LTOT * 64) + l * 64 + col];
      s_x1[item * LTOT + l][col] = v;
      atomicAdd(&s_ss[item][d], v * v);
    });
  __syncthreads();
  if (threadIdx.x < 10) {
    int item = threadIdx.x / 5, d = threadIdx.x % 5;
    float cnt = (float)((2 * d + 1) * CCH);
    s_inv[item][d] = rsqrtf(s_ss[item][d] / cnt + EPSV);
  }
  __syncthreads();
  for (int idx = threadIdx.x; idx < 2 * LTOT * 64; idx += blockDim.x) {
    int rowl = idx >> 6, c = idx & 63;
    int item = rowl / LTOT, l = rowl % LTOT;
    int d = deg_of(l);
    float v = s_x1[rowl][c];
    x1[(size_t)n0 * (LTOT * 64) + idx] = v;
    yn[(size_t)n0 * (LTOT * 64) + idx] = v * s_inv[item][d] * n2w[d * 64 + c];
  }
}

// ---------------------------------------------------------------------------
// K8: grid FFN, one node per block, 5 chained WMMA GEMMs, phase-overlapped LDS
// pool:  g @ [0,5940), g2 @ [5940,11880), yn @ [5940,7640), y1 @ [7640,10940),
// y2 @ [0,3300)
// ---------------------------------------------------------------------------
__global__ void k_node_ffn(const float* __restrict__ yn_g, const float* __restrict__ x1,
                           const float* __restrict__ W_ff1, const float* __restrict__ to_grid,
                           const float* __restrict__ from_grid, const float* __restrict__ W_grid,
                           const float* __restrict__ b_grid, const float* __restrict__ W_ff2,
                           float* __restrict__ out) {
  __shared__ float sm[11904];
  float* s_g  = sm;           // [45][132]
  float* s_g2 = sm + 5940;    // [45][132]
  float* s_yn = sm + 5940;    // [25][68]  (dead before s_g2 is live)
  float* s_y1 = sm + 7640;    // [25][132] (dead before s_g2 is live)
  float* s_y2 = sm;           // [25][132] (overwrites s_g after it is dead)
  const int n = blockIdx.x;
  const int wid = threadIdx.x >> 5, nw = blockDim.x >> 5;
  for (int idx = threadIdx.x; idx < LTOT * 64; idx += blockDim.x)
    s_yn[(idx >> 6) * 68 + (idx & 63)] = yn_g[(size_t)n * (LTOT * 64) + idx];
  __syncthreads();
  // y1 = deg_linear(yn, W_ff1): K=64 -> N=128
  gemm_tiles(5, 8, 16, wid, nw,
    [&](int row, int k) {
      int d = row >> 4, lm = row & 15;
      return (lm <= 2 * d) ? s_yn[(d * d + lm) * 68 + k] : 0.f;
    },
    [&](int mt, int k, int col) { return W_ff1[(size_t)mt * (64 * 128) + k * 128 + col]; },
    [&](int, int row, int col, float acc) {
      int d = row >> 4, lm = row & 15;
      if (lm <= 2 * d) s_y1[(d * d + lm) * 132 + col] = acc;
    });
  __syncthreads();
  // g = to_grid @ y1: M=45, K=25 -> N=128
  gemm_tiles(3, 8, 7, wid, nw,
    [&](int row, int k) { return (row < 45 && k < 25) ? to_grid[row * 25 + k] : 0.f; },
    [&](int, int k, int col) { return (k < 25) ? s_y1[k * 132 + col] : 0.f; },
    [&](int, int row, int col, float acc) { if (row < 45) s_g[row * 132 + col] = acc; });
  __syncthreads();
  // g2 = silu(g @ W_grid + b_grid): M=45, K=128 -> N=128
  gemm_tiles(3, 8, 32, wid, nw,
    [&](int row, int k) { return (row < 45) ? s_g[row * 132 + k] : 0.f; },
    [&](int, int k, int col) { return W_grid[k * 128 + col]; },
    [&](int, int row, int col, float acc) {
      if (row < 45) s_g2[row * 132 + col] = siluf(acc + b_grid[col]);
    });
  __syncthreads();
  // y2 = from_grid @ g2: M=25, K=45 -> N=128
  gemm_tiles(2, 8, 12, wid, nw,
    [&](int row, int k) { return (row < 25 && k < 45) ? from_grid[row * 45 + k] : 0.f; },
    [&](int, int k, int col) { return (k < 45) ? s_g2[k * 132 + col] : 0.f; },
    [&](int, int row, int col, float acc) { if (row < 25) s_y2[row * 132 + col] = acc; });
  __syncthreads();
  // out = deg_linear(y2, W_ff2) + x1: K=128 -> N=64
  gemm_tiles(5, 4, 32, wid, nw,
    [&](int row, int k) {
      int d = row >> 4, lm = row & 15;
      return (lm <= 2 * d) ? s_y2[(d * d + lm) * 132 + k] : 0.f;
    },
    [&](int mt, int k, int col) { return W_ff2[(size_t)mt * (128 * 64) + k * 64 + col]; },
    [&](int, int row, int col, float acc) {
      int d = row >> 4, lm = row & 15;
      if (lm <= 2 * d) {
        size_t o = (size_t)n * (LTOT * 64) + (d * d + lm) * 64 + col;
        out[o] = acc + x1[o];
      }
    });
}

// ---------------------------------------------------------------------------
extern "C" void kernel_launch(void* const* d_in, const int* in_sizes, int n_in,
                              void* d_out, int out_size, void* d_ws, size_t ws_size,
                              hipStream_t stream) {
  const float* node_feats = (const float*)d_in[0];
  const int*   species    = (const int*)d_in[1];
  const float* dist       = (const float*)d_in[2];
  const int*   snd        = (const int*)d_in[3];
  const int*   rcv        = (const int*)d_in[4];
  const float* wig        = (const float*)d_in[5];
  const float* n1w        = (const float*)d_in[7];
  const float* n2w        = (const float*)d_in[8];
  const float* W_edge     = (const float*)d_in[9];
  const float* emb_s      = (const float*)d_in[10];
  const float* emb_d      = (const float*)d_in[11];
  const float* W_r1       = (const float*)d_in[12];
  const float* b_r1       = (const float*)d_in[13];
  const float* W_r2       = (const float*)d_in[14];
  const float* b_r2       = (const float*)d_in[15];
  const float* W_msg      = (const float*)d_in[16];
  const float* W_alpha    = (const float*)d_in[17];
  const float* alpha_vec  = (const float*)d_in[18];
  const float* W_val      = (const float*)d_in[19];
  const float* W_out      = (const float*)d_in[20];
  const float* W_ff1      = (const float*)d_in[21];
  const float* to_grid    = (const float*)d_in[22];
  const float* from_grid  = (const float*)d_in[23];
  const float* W_grid     = (const float*)d_in[24];
  const float* b_grid     = (const float*)d_in[25];
  const float* W_ff2      = (const float*)d_in[26];

  const int N = in_sizes[0] / (LTOT * CCH);
  const int E = in_sizes[2] / 64;

  float* ws = (float*)d_ws;
  size_t off = 0;
  float* xn      = ws + off; off += (size_t)N * LTOT * 64;
  float* rad     = ws + off; off += (size_t)E * 64;
  float* hbuf    = ws + off; off += (size_t)E * LTOT * 64;
  float* logits  = ws + off; off += (size_t)E * NHEADS;
  float* exb     = ws + off; off += (size_t)E * NHEADS;
  unsigned* nmax = (unsigned*)(ws + off); off += (size_t)N * NHEADS;
  float* denom   = ws + off; off += (size_t)N * NHEADS;
  float* agg     = ws + off; off += (size_t)N * LTOT * 128;
  float* x1      = ws + off; off += (size_t)N * LTOT * 64;
  float* yn      = ws + off; off += (size_t)N * LTOT * 64;

  k_norm<<<N, 256, 0, stream>>>(node_feats, n1w, xn);
  k_radial<<<E / 64, 128, 0, stream>>>(dist, species, snd, rcv, W_edge, emb_s, emb_d,
                                       W_r1, b_r1, W_r2, b_r2, rad);
  k_init<<<1024, 256, 0, stream>>>(agg, denom, nmax, N * LTOT * 128, N * NHEADS);
  k_edge_msg<<<E, 256, 0, stream>>>(xn, snd, rcv, wig, rad, W_msg, W_alpha, alpha_vec,
                                    hbuf, logits);
  k_attn_max<<<(E * NHEADS + 255) / 256, 256, 0, stream>>>(logits, rcv, nmax, E);
  k_attn_sum<<<(E * NHEADS + 255) / 256, 256, 0, stream>>>(logits, rcv, nmax, exb, denom, E);
  k_edge_val<<<E, 256, 0, stream>>>(hbuf, rcv, wig, W_val, exb, denom, agg);
  k_node_out<<<N / 2, 256, 0, stream>>>(agg, node_feats, W_out, n2w, x1, yn);
  k_node_ffn<<<N, 256, 0, stream>>>(yn, x1, W_ff1, to_grid, from_grid, W_grid, b_grid,
                                    W_ff2, (float*)d_out);
}